// _SurfaceSDF_86414741996099
// MI455X (gfx1250) — compile-verified
//
#include <hip/hip_runtime.h>
#include <math.h>

typedef __attribute__((ext_vector_type(2))) float v2f;
typedef __attribute__((ext_vector_type(8))) float v8f;

#define NX 16384
#define NA 4096
#define FEPS 1e-12f

__global__ __launch_bounds__(256)
void _SurfaceSDF_kernel(const float* __restrict__ X,
                        const float* __restrict__ A,
                        const float* __restrict__ SIG,
                        float* __restrict__ out) {
    const int tid  = threadIdx.x;
    const int wave = tid >> 5;
    const int lane = tid & 31;
    const int half = lane >> 4;      // 0 = lanes 0-15, 1 = lanes 16-31
    const int l16  = lane & 15;

    const int rowBase = blockIdx.x * 128 + wave * 16;  // 16 query rows per wave
    const int mload   = rowBase + l16;

    // Load this lane's query point (row-major [NX,3])
    const float xx = X[mload * 3 + 0];
    const float xy = X[mload * 3 + 1];
    const float xz = X[mload * 3 + 2];
    const float x2full = xx * xx + xy * xy + xz * xz;

    // A-operand (16x4 f32): lanes 0-15 hold K=0,1 (x,y); lanes 16-31 hold K=2,3 (z,pad)
    v2f aop;
    aop.x = half ? xz : xx;
    aop.y = half ? 0.f : xy;

    // |x|^2 for the 8 rows this lane's C/D accumulators map to (row = r + 8*half)
    float x2v[8];
#pragma unroll
    for (int r = 0; r < 8; ++r)
        x2v[r] = __shfl(x2full, r + half * 8, 32);

    float numAcc[8], denAcc[8], mAcc[8], sAcc[8];
#pragma unroll
    for (int r = 0; r < 8; ++r) {
        numAcc[r] = 0.f; denAcc[r] = 0.f;
        mAcc[r]   = -INFINITY; sAcc[r] = 0.f;
    }

    // Loop over 256 tiles of 16 atoms; x2 unroll keeps two WMMAs in flight
#pragma unroll 2
    for (int t = 0; t < NA / 16; ++t) {
        const int n = t * 16 + l16;
        const float ax = A[n * 3 + 0];
        const float ay = A[n * 3 + 1];
        const float az = A[n * 3 + 2];
        const float sg  = SIG[n];
        const float inv = __builtin_amdgcn_rcpf(sg + FEPS);  // v_rcp_f32
        const float a2  = ax * ax + ay * ay + az * az;

        // B-operand (4x16 f32): column n = l16; same K split as A-operand
        v2f bop;
        bop.x = half ? az : ax;
        bop.y = half ? 0.f : ay;

        // dot(x_m, a_n) for the 16x16 tile
        v8f c = {};
        c = __builtin_amdgcn_wmma_f32_16x16x4_f32(
                /*neg_a=*/false, aop, /*neg_b=*/false, bop,
                /*c_mod=*/(short)0, c, /*reuse_a=*/false, /*reuse_b=*/false);

#pragma unroll
        for (int r = 0; r < 8; ++r) {
            float d2 = x2v[r] + a2 - 2.f * c[r];
            float d  = __builtin_amdgcn_sqrtf(fmaxf(d2, 0.f));  // v_sqrt_f32
            float e  = __expf(-d);
            denAcc[r] += e;
            numAcc[r]  = fmaf(e, sg, numAcc[r]);
            // online logsumexp of v = -d/sigma
            float v  = -d * inv;
            float mo = mAcc[r];
            float mn = fmaxf(mo, v);
            sAcc[r]  = sAcc[r] * __expf(mo - mn) + __expf(v - mn);
            mAcc[r]  = mn;
        }
    }

    // Reduce over the 16 atom-columns (lanes within each half; masks stay in-half)
#pragma unroll
    for (int r = 0; r < 8; ++r) {
        float num = numAcc[r], den = denAcc[r], M = mAcc[r], S = sAcc[r];
#pragma unroll
        for (int mask = 1; mask < 16; mask <<= 1) {
            num += __shfl_xor(num, mask, 32);
            den += __shfl_xor(den, mask, 32);
            float oM = __shfl_xor(M, mask, 32);
            float oS = __shfl_xor(S, mask, 32);
            float nM = fmaxf(M, oM);
            S = S * __expf(M - nM) + oS * __expf(oM - nM);
            M = nM;
        }
        if (l16 == 0) {
            float f   = num / (den + FEPS);   // cold path: keep precise divide
            float lse = M + __logf(S);
            out[rowBase + r + half * 8] = -f * lse;
        }
    }
}

extern "C" void kernel_launch(void* const* d_in, const int* in_sizes, int n_in,
                              void* d_out, int out_size, void* d_ws, size_t ws_size,
                              hipStream_t stream) {
    const float* X   = (const float*)d_in[0];
    const float* A   = (const float*)d_in[1];
    const float* SIG = (const float*)d_in[2];
    float* out = (float*)d_out;

    dim3 grid(NX / 128);   // 128 rows per block (8 waves x 16 rows)
    dim3 block(256);
    hipLaunchKernelGGL(_SurfaceSDF_kernel, grid, block, 0, stream, X, A, SIG, out);
}